// DCNv2Pooling_28424093565278
// MI455X (gfx1250) — compile-verified
//
#include <hip/hip_runtime.h>
#include <hip/hip_bf16.h>
#include <cstdint>

// ---------------------------------------------------------------------------
// DCNv2 deformable PS-ROI pooling for MI455X (gfx1250).
// Pass 1: NCHW -> NHWC transpose into workspace (coalesced tiled transpose).
// Pass 2: one workgroup per ROI; 4 subgroups of 64 threads each own one bin
//         at a time; each thread handles 4 channels with float4 NHWC loads.
//         Result tile (256 ch x 49 bins) staged in LDS, then streamed to
//         global with the Tensor Data Mover (tensor_store_from_lds).
// ---------------------------------------------------------------------------

#define B_   4
#define C_   256
#define H_   128
#define W_   128
#define PB   7            // POOLED bins per side
#define NBIN 49
#define SPATIAL_SCALE 0.0625f
#define TRANS_STD 0.1f

typedef unsigned int u32x4 __attribute__((ext_vector_type(4)));
typedef int          i32x4 __attribute__((ext_vector_type(4)));
typedef int          i32x8 __attribute__((ext_vector_type(8)));

#if __has_builtin(__builtin_amdgcn_tensor_store_from_lds)
#define HAVE_TDM 1
#endif
#if __has_include(<hip/amd_detail/amd_gfx1250_TDM.h>)
#define TDM_6ARG 1
#endif

__device__ __forceinline__ float4 ld4(const float* p) {
    return *reinterpret_cast<const float4*>(p);
}

template<bool NHWC>
__device__ __forceinline__ float4 gather4(const float* __restrict__ f,
                                          int b, int y, int x, int c0) {
    if (NHWC) {
        return ld4(f + (((size_t)(b * H_ + y) * W_ + x) * C_ + c0));
    } else {
        const size_t HW = (size_t)H_ * W_;
        const float* p = f + (((size_t)(b * C_ + c0) * H_ + y) * W_ + x);
        float4 r;
        r.x = p[0];
        r.y = p[HW];
        r.z = p[2 * HW];
        r.w = p[3 * HW];
        return r;
    }
}

// ---------------------------------------------------------------------------
// Pass 1: NCHW -> NHWC transpose.  View per-batch as (C, H*W) -> (H*W, C).
// grid = (HW/32, C/32, B), block = 256 (32x8), LDS tile 32x33 (conflict-free).
// ---------------------------------------------------------------------------
__global__ __launch_bounds__(256)
void nchw_to_nhwc(const float* __restrict__ in, float* __restrict__ ws) {
    __shared__ float t[32][33];
    const int x   = threadIdx.x & 31;
    const int y   = threadIdx.x >> 5;
    const int hw0 = blockIdx.x * 32;
    const int cb0 = blockIdx.y * 32;
    const int b   = blockIdx.z;
    const size_t HW = (size_t)H_ * W_;
#pragma unroll
    for (int k = 0; k < 4; ++k) {
        int c = cb0 + y + k * 8;
        t[y + k * 8][x] = in[((size_t)b * C_ + c) * HW + hw0 + x];
    }
    __syncthreads();
#pragma unroll
    for (int k = 0; k < 4; ++k) {
        int hw = hw0 + y + k * 8;
        ws[((size_t)b * HW + hw) * C_ + cb0 + x] = t[x][y + k * 8];
    }
}

// ---------------------------------------------------------------------------
// Pass 2: deformable PS-ROI pooling, one block (256 thr = 8 waves) per ROI.
// subgroup = tid>>6 (a whole pair of wave32s -> uniform branches per wave),
// each thread owns 4 channels (float4 path when NHWC).
// ---------------------------------------------------------------------------
template<bool NHWC>
__global__ __launch_bounds__(256)
void psroi_kernel(const float* __restrict__ feat,
                  const float* __restrict__ rois,
                  const float* __restrict__ offset,
                  float* __restrict__ out) {
    __shared__ float tile[C_ * NBIN];   // [c][bin], 50176 B

    const int n   = blockIdx.x;
    const int tid = threadIdx.x;
    const int sub = tid >> 6;
    const int t   = tid & 63;
    const int c0  = t * 4;

    const float* roi = rois + (size_t)n * 5;
    const int   b  = (int)roi[0];
    const float x1 = rintf(roi[1]) * SPATIAL_SCALE - 0.5f;
    const float y1 = rintf(roi[2]) * SPATIAL_SCALE - 0.5f;
    const float x2 = (rintf(roi[3]) + 1.0f) * SPATIAL_SCALE - 0.5f;
    const float y2 = (rintf(roi[4]) + 1.0f) * SPATIAL_SCALE - 0.5f;
    const float rw = fmaxf(x2 - x1, 0.1f);
    const float rh = fmaxf(y2 - y1, 0.1f);
    const float bw = rw / (float)PB;
    const float bh = rh / (float)PB;
    const float sw = bw / 4.0f;
    const float sh = bh / 4.0f;

    for (int bin = sub; bin < NBIN; bin += 4) {
        const int ph = bin / PB;
        const int pw = bin - ph * PB;
        const float tx = offset[((size_t)n * 2 + 0) * NBIN + bin] * TRANS_STD;
        const float ty = offset[((size_t)n * 2 + 1) * NBIN + bin] * TRANS_STD;
        const float wstart = (float)pw * bw + x1 + tx * rw;
        const float hstart = (float)ph * bh + y1 + ty * rh;

        if (NHWC) {
            // Prefetch the far corner of this bin's footprint (gfx1250
            // global_prefetch_b8); the sample loop walks toward it.
            float wl = fminf(fmaxf(wstart + 3.0f * sw, 0.0f), (float)W_ - 1.0f);
            float hl = fminf(fmaxf(hstart + 3.0f * sh, 0.0f), (float)H_ - 1.0f);
            int xx = (int)wl, yy = (int)hl;
            int yyn = (yy + 1 < H_) ? yy + 1 : H_ - 1;
            __builtin_prefetch(feat + ((size_t)(b * H_ + yy ) * W_ + xx) * C_ + c0, 0, 0);
            __builtin_prefetch(feat + ((size_t)(b * H_ + yyn) * W_ + xx) * C_ + c0, 0, 0);
        }

        float4 acc = make_float4(0.f, 0.f, 0.f, 0.f);
        float  cnt = 0.f;

#pragma unroll
        for (int s = 0; s < 16; ++s) {          // ih outer, iw inner (ref order)
            const int   ih = s >> 2;
            const int   iw = s & 3;
            const float w  = wstart + (float)iw * sw;
            const float h  = hstart + (float)ih * sh;
            const bool valid = (w >= -0.5f) && (w <= (float)W_ - 0.5f) &&
                               (h >= -0.5f) && (h <= (float)H_ - 0.5f);
            if (!valid) continue;               // wave-uniform branch

            const float wc  = fminf(fmaxf(w, 0.0f), (float)W_ - 1.0f);
            const float hc  = fminf(fmaxf(h, 0.0f), (float)H_ - 1.0f);
            const float x0f = floorf(wc);
            const float y0f = floorf(hc);
            const int   x0  = (int)x0f;
            const int   y0  = (int)y0f;
            const int   x1i = (int)ceilf(wc);   // matches jnp.ceil gather
            const int   y1i = (int)ceilf(hc);
            const float dx  = wc - x0f;
            const float dy  = hc - y0f;
            const float odx = 1.0f - dx;
            const float ody = 1.0f - dy;

            const float4 v00 = gather4<NHWC>(feat, b, y0,  x0,  c0);
            const float4 v01 = gather4<NHWC>(feat, b, y0,  x1i, c0);
            const float4 v10 = gather4<NHWC>(feat, b, y1i, x0,  c0);
            const float4 v11 = gather4<NHWC>(feat, b, y1i, x1i, c0);

            acc.x += ody * (odx * v00.x + dx * v01.x) + dy * (odx * v10.x + dx * v11.x);
            acc.y += ody * (odx * v00.y + dx * v01.y) + dy * (odx * v10.y + dx * v11.y);
            acc.z += ody * (odx * v00.z + dx * v01.z) + dy * (odx * v10.z + dx * v11.z);
            acc.w += ody * (odx * v00.w + dx * v01.w) + dy * (odx * v10.w + dx * v11.w);
            cnt   += 1.0f;
        }

        float4 o;
        if (cnt > 0.0f) {
            o.x = acc.x / cnt; o.y = acc.y / cnt;
            o.z = acc.z / cnt; o.w = acc.w / cnt;
        } else {
            o = make_float4(0.f, 0.f, 0.f, 0.f);
        }
        tile[(c0 + 0) * NBIN + bin] = o.x;
        tile[(c0 + 1) * NBIN + bin] = o.y;
        tile[(c0 + 2) * NBIN + bin] = o.z;
        tile[(c0 + 3) * NBIN + bin] = o.w;
    }

    __syncthreads();

    // ---- Store the 49 KB result tile: out[n, c, ph, pw] = tile[c*49 + bin].
#ifdef HAVE_TDM
    if (tid < 32) {   // single wave issues one TDM descriptor per ROI
        const uint32_t lds = (uint32_t)(uintptr_t)(void*)&tile[0];
        const uint64_t ga  = (uint64_t)(uintptr_t)(out + (size_t)n * C_ * NBIN);
        // D# group0: count=1, lds_addr, global_addr, type=2 ("image")
        u32x4 g0;
        g0[0] = 1u;
        g0[1] = lds;
        g0[2] = (uint32_t)(ga & 0xFFFFFFFFu);
        g0[3] = ((uint32_t)(ga >> 32) & 0x01FFFFFFu) | (2u << 30);
        // D# group1: data_size=4B; 2D tile x=bin(49), y=c(256); stride0=49.
        i32x8 g1;
        g1[0] = 2 << 16;            // data_size = 2 (4 bytes)
        g1[1] = NBIN << 16;         // tensor_dim0 lo16 = 49
        g1[2] = C_ << 16;           // tensor_dim0 hi16=0 | tensor_dim1 lo16=256
        g1[3] = NBIN << 16;         // tensor_dim1 hi16=0 | tile_dim0 = 49
        g1[4] = C_;                 // tile_dim1 = 256, tile_dim2 = 0
        g1[5] = NBIN;               // tensor_dim0_stride = 49
        g1[6] = 0;                  // stride0 hi | stride1 lo
        g1[7] = 0;                  // stride1 hi
        i32x4 z4 = {0, 0, 0, 0};
#ifdef TDM_6ARG
        i32x8 z8 = {0, 0, 0, 0, 0, 0, 0, 0};
        __builtin_amdgcn_tensor_store_from_lds(g0, g1, z4, z4, z8, 0);
#else
        __builtin_amdgcn_tensor_store_from_lds(g0, g1, z4, z4, 0);
#endif
#if __has_builtin(__builtin_amdgcn_s_wait_tensorcnt)
        __builtin_amdgcn_s_wait_tensorcnt(0);
#endif
    }
#else
    float* dst = out + (size_t)n * C_ * NBIN;
    for (int i = tid; i < C_ * NBIN; i += 256) dst[i] = tile[i];
#endif
}

// ---------------------------------------------------------------------------
extern "C" void kernel_launch(void* const* d_in, const int* in_sizes, int n_in,
                              void* d_out, int out_size, void* d_ws, size_t ws_size,
                              hipStream_t stream) {
    const float* inp  = (const float*)d_in[0];
    const float* rois = (const float*)d_in[1];
    const float* off  = (const float*)d_in[2];
    float*       out  = (float*)d_out;

    const int N = in_sizes[1] / 5;
    const size_t need = (size_t)B_ * C_ * H_ * W_ * sizeof(float);

    if (ws_size >= need) {
        float* ws = (float*)d_ws;
        dim3 tgrid((H_ * W_) / 32, C_ / 32, B_);
        nchw_to_nhwc<<<tgrid, 256, 0, stream>>>(inp, ws);
        psroi_kernel<true><<<N, 256, 0, stream>>>(ws, rois, off, out);
    } else {
        psroi_kernel<false><<<N, 256, 0, stream>>>(inp, rois, off, out);
    }
}